// GCN_Conv_Sum_35888746725385
// MI455X (gfx1250) — compile-verified
//
#include <hip/hip_runtime.h>
#include <hip/hip_bf16.h>
#include <math.h>

typedef float v2f __attribute__((ext_vector_type(2)));
typedef float v8f __attribute__((ext_vector_type(8)));

#define THREADS 256

// ---------------------------------------------------------------------------
// deg / init kernels
// ---------------------------------------------------------------------------
__global__ void k_init_one(float* __restrict__ p, int n) {
  int i = blockIdx.x * THREADS + threadIdx.x;
  if (i < n) p[i] = 1.0f;                    // self-loop contributes 1 to deg
}

__global__ void k_zero(float* __restrict__ p, long n) {
  long i = (long)blockIdx.x * THREADS + threadIdx.x;
  if (i < n) p[i] = 0.0f;
}

__global__ void k_edge_deg(const int* __restrict__ dst, float* __restrict__ deg, int e) {
  int i = blockIdx.x * THREADS + threadIdx.x;
  if (i < e) atomicAdd(&deg[dst[i]], 1.0f);
}

__global__ void k_rsqrt_inplace(float* __restrict__ d, int n) {
  int i = blockIdx.x * THREADS + threadIdx.x;
  if (i < n) d[i] = rsqrtf(d[i]);            // deg >= 1 always (self-loops)
}

// ---------------------------------------------------------------------------
// Weight prep: WT[c][k] = (c < 8) ? W[k][c] : 0   for c in [0,16), k in [0,K)
// Transposed + zero-padded so the GEMM's B-fragment loads are unconditional
// contiguous b64 loads (no exec-masked branches in the hot loop).
// Launch with exactly 16*K threads, 1 block.
// ---------------------------------------------------------------------------
__global__ void k_pad_wT(const float* __restrict__ W, float* __restrict__ WT, int K) {
  int i = threadIdx.x;          // i = c*K + k
  int c = i / K;
  int k = i % K;
  WT[i] = (c < 8) ? W[k * 8 + c] : 0.f;
}

// ---------------------------------------------------------------------------
// Dense transform  H[N,8] = X[N,K] @ W[K,8]   via V_WMMA_F32_16X16X4_F32.
// One wave computes a 16-row tile; K is compile-time (16 or 8) -> full unroll.
//   A (32-bit 16x4): lane L -> row (L&15), holds K = (L>>4)*2 and +1
//   B (32-bit 4x16): lane L -> col (L&15), holds K = (L>>4)*2 and +1
//   C/D (32-bit 16x16): VGPR r -> row (lane>>4)*8 + r, col = lane&15
// ---------------------------------------------------------------------------
template <int K>
__global__ __launch_bounds__(THREADS)
void k_gemm_wmma(const float* __restrict__ X, const float* __restrict__ WT,
                 float* __restrict__ H, int nrows) {
  const int lane  = threadIdx.x & 31;
  const int wave  = threadIdx.x >> 5;
  const long m0   = ((long)blockIdx.x * 8 + wave) * 16;   // 8 waves / block
  if (m0 >= nrows) return;                                // wave-uniform exit

  const int mrow = lane & 15;
  const int kg   = (lane >> 4) * 2;                       // K sub-pair: 0 or 2
  const int col  = lane & 15;

  long row = m0 + mrow;
  long rld = (row < nrows) ? row : (long)(nrows - 1);     // clamp (extra rows unstored)

  const float* xp = X + rld * K + kg;                     // A fragment base
  const float* wp = WT + (long)col * K + kg;              // B fragment base (padded/T)

  v2f a[K / 4];
  v2f b[K / 4];
#pragma unroll
  for (int s = 0; s < K / 4; ++s) {
    a[s] = *(const v2f*)(xp + s * 4);                     // global_load_b64
    b[s] = *(const v2f*)(wp + s * 4);                     // global_load_b64
  }

  v8f acc = {0.f, 0.f, 0.f, 0.f, 0.f, 0.f, 0.f, 0.f};
#pragma unroll
  for (int s = 0; s < K / 4; ++s) {
    acc = __builtin_amdgcn_wmma_f32_16x16x4_f32(
        /*neg_a=*/false, a[s], /*neg_b=*/false, b[s],
        /*c_mod=*/(short)0, acc, /*reuse_a=*/false, /*reuse_b=*/false);
  }

  const int mbase = (lane >> 4) * 8;
  if (col < 8) {
    if (m0 + 16 <= nrows) {                               // fast path: full tile
#pragma unroll
      for (int r = 0; r < 8; ++r) H[(m0 + mbase + r) * 8 + col] = acc[r];
    } else {
#pragma unroll
      for (int r = 0; r < 8; ++r) {
        long orow = m0 + mbase + r;
        if (orow < nrows) H[orow * 8 + col] = acc[r];
      }
    }
  }
}

// ---------------------------------------------------------------------------
// Edge scatter: agg[dst] += dis[src]*dis[dst] * h[src]   (one thread / edge)
// ---------------------------------------------------------------------------
__global__ void k_scatter(const float* __restrict__ h, const int* __restrict__ src,
                          const int* __restrict__ dst, const float* __restrict__ dis,
                          float* __restrict__ agg, int e) {
  int i = blockIdx.x * THREADS + threadIdx.x;
  if (i >= e) return;
  int s = src[i];
  int d = dst[i];
  float w = dis[s] * dis[d];
  const float4* hp = (const float4*)(h + (long)s * 8);
  float4 h0 = hp[0];
  float4 h1 = hp[1];
  float* o = agg + (long)d * 8;
  atomicAdd(o + 0, h0.x * w);
  atomicAdd(o + 1, h0.y * w);
  atomicAdd(o + 2, h0.z * w);
  atomicAdd(o + 3, h0.w * w);
  atomicAdd(o + 4, h1.x * w);
  atomicAdd(o + 5, h1.y * w);
  atomicAdd(o + 6, h1.z * w);
  atomicAdd(o + 7, h1.w * w);
}

// ---------------------------------------------------------------------------
// Epilogue: agg = relu(agg + h*dis^2 + b)   (self-loop folded in, no atomics)
// ---------------------------------------------------------------------------
__global__ void k_selfloop_bias_relu(float* __restrict__ agg, const float* __restrict__ h,
                                     const float* __restrict__ dis,
                                     const float* __restrict__ b, int n) {
  long i = (long)blockIdx.x * THREADS + threadIdx.x;
  if (i >= (long)n * 8) return;
  int node = (int)(i >> 3);
  int f    = (int)(i & 7);
  float ds = dis[node];
  float v  = agg[i] + h[i] * ds * ds + b[f];
  agg[i] = v > 0.f ? v : 0.f;
}

// ---------------------------------------------------------------------------
// Global add pool: pooled[batch[node]][f] += x[node][f]
// ---------------------------------------------------------------------------
__global__ void k_pool(const float* __restrict__ x, const int* __restrict__ batch,
                       float* __restrict__ pooled, int n) {
  long i = (long)blockIdx.x * THREADS + threadIdx.x;
  if (i >= (long)n * 8) return;
  int node = (int)(i >> 3);
  int f    = (int)(i & 7);
  atomicAdd(&pooled[(long)batch[node] * 8 + f], x[i]);
}

// ---------------------------------------------------------------------------
// Head: out[g] = sigmoid(pooled[g,:] . Wl + bl)
// ---------------------------------------------------------------------------
__global__ void k_head(const float* __restrict__ pooled, const float* __restrict__ Wl,
                       const float* __restrict__ bl, float* __restrict__ out, int g) {
  int i = blockIdx.x * THREADS + threadIdx.x;
  if (i >= g) return;
  float z = bl[0];
#pragma unroll
  for (int f = 0; f < 8; ++f) z += pooled[(long)i * 8 + f] * Wl[f];
  out[i] = 1.0f / (1.0f + expf(-z));
}

// ---------------------------------------------------------------------------
extern "C" void kernel_launch(void* const* d_in, const int* in_sizes, int n_in,
                              void* d_out, int out_size, void* d_ws, size_t ws_size,
                              hipStream_t stream) {
  const float* x  = (const float*)d_in[0];   // [N,16]
  const int*   ei = (const int*)d_in[1];     // [2,E] : row0 = src, row1 = dst
  const int*   bt = (const int*)d_in[2];     // [N]
  const float* W1 = (const float*)d_in[3];   // [16,8]
  const float* b1 = (const float*)d_in[4];   // [8]
  const float* W2 = (const float*)d_in[5];   // [8,8]
  const float* b2 = (const float*)d_in[6];   // [8]
  const float* Wl = (const float*)d_in[7];   // [8,1]
  const float* bl = (const float*)d_in[8];   // [1]

  const int N = in_sizes[0] / 16;
  const int E = in_sizes[1] / 2;
  const int G = out_size;                    // 512

  const int* src = ei;
  const int* dst = ei + E;

  // Workspace (floats):
  //   dis[N] | h[N*8] | aggA[N*8] | aggB[N*8] | pooled[G*8] | WT1[256] | WT2[128]
  float* dis    = (float*)d_ws;
  float* h      = dis + N;
  float* aggA   = h + (long)N * 8;
  float* aggB   = aggA + (long)N * 8;
  float* pooled = aggB + (long)N * 8;
  float* WT1    = pooled + (long)G * 8;
  float* WT2    = WT1 + 16 * 16;

  const int gN   = (N + THREADS - 1) / THREADS;
  const int gE   = (E + THREADS - 1) / THREADS;
  const int gN8  = (int)(((long)N * 8 + THREADS - 1) / THREADS);
  const int gGem = (int)((((long)N + 15) / 16 + 7) / 8);   // 8 waves (16 rows) / block

  // --- degree / symmetric normalization + weight prep ---------------------
  k_init_one<<<gN, THREADS, 0, stream>>>(dis, N);
  k_edge_deg<<<gE, THREADS, 0, stream>>>(dst, dis, E);
  k_rsqrt_inplace<<<gN, THREADS, 0, stream>>>(dis, N);
  k_pad_wT<<<1, 16 * 16, 0, stream>>>(W1, WT1, 16);
  k_pad_wT<<<1, 16 * 8, 0, stream>>>(W2, WT2, 8);

  // --- layer 1: h = x @ W1 ; scatter ; relu(agg + h*dis^2 + b1) -----------
  k_gemm_wmma<16><<<gGem, THREADS, 0, stream>>>(x, WT1, h, N);
  k_zero<<<gN8, THREADS, 0, stream>>>(aggA, (long)N * 8);
  k_scatter<<<gE, THREADS, 0, stream>>>(h, src, dst, dis, aggA, E);
  k_selfloop_bias_relu<<<gN8, THREADS, 0, stream>>>(aggA, h, dis, b1, N);

  // --- layer 2: h = aggA @ W2 ; scatter ; relu(agg + h*dis^2 + b2) --------
  k_gemm_wmma<8><<<gGem, THREADS, 0, stream>>>(aggA, WT2, h, N);
  k_zero<<<gN8, THREADS, 0, stream>>>(aggB, (long)N * 8);
  k_scatter<<<gE, THREADS, 0, stream>>>(h, src, dst, dis, aggB, E);
  k_selfloop_bias_relu<<<gN8, THREADS, 0, stream>>>(aggB, h, dis, b2, N);

  // --- pool + head --------------------------------------------------------
  k_zero<<<(G * 8 + THREADS - 1) / THREADS, THREADS, 0, stream>>>(pooled, (long)G * 8);
  k_pool<<<gN8, THREADS, 0, stream>>>(aggB, bt, pooled, N);
  k_head<<<(G + THREADS - 1) / THREADS, THREADS, 0, stream>>>(pooled, Wl, bl, (float*)d_out, G);
}